// RepulsionLoss_1992864825913
// MI455X (gfx1250) — compile-verified
//
#include <hip/hip_runtime.h>
#include <hip/hip_bf16.h>
#include <math.h>

#define BATCH   64
#define NPRIOR  24564
#define NGT     48
#define THRESH  0.5f
#define VAR0    0.1f
#define VAR1    0.2f
#define SIGMA   0.5f
#define IOG_CAP 0.95f
#define EPSV    1e-7f
#define PTILE   256

// Workspace layout (d_ws):
//   float acc[0] = loss_sum, acc[1] = pos_count   (zeroed every call by k_init)
//   int   best_prior_idx[BATCH*NGT] at byte offset 16 (fully overwritten every call)

__device__ __forceinline__ float iou_corner(float ax1, float ay1, float ax2, float ay2, float areaA,
                                            float bx1, float by1, float bx2, float by2, float areaB) {
    float lx = fmaxf(ax1, bx1), ly = fmaxf(ay1, by1);
    float rx = fminf(ax2, bx2), ry = fminf(ay2, by2);
    float w = fmaxf(rx - lx, 0.0f), h = fmaxf(ry - ly, 0.0f);
    float inter = w * h;
    return inter / (areaA + areaB - inter);
}

__global__ void k_init(float* acc) {
    if (threadIdx.x < 2) acc[threadIdx.x] = 0.0f;
}

// One block per (b,g): best_prior_idx[b,g] = argmax_p IoU(truths[b,g], point_form(priors[p]))
// Tie-break: lowest p (matches jnp.argmax first-occurrence).
__global__ void k_best_prior(const float* __restrict__ priors,
                             const float* __restrict__ gt,   // [B, G, 5]
                             int* __restrict__ bp) {
    int b = blockIdx.x / NGT, g = blockIdx.x % NGT;
    const float* gtb = gt + ((size_t)b * NGT + g) * 5;
    float gx1 = gtb[0], gy1 = gtb[1], gx2 = gtb[2], gy2 = gtb[3];
    float areaG = (gx2 - gx1) * (gy2 - gy1);

    float best = -1.0f;
    int bidx = 0x7fffffff;
    for (int p = threadIdx.x; p < NPRIOR; p += blockDim.x) {
        const float* pr = priors + (size_t)p * 4;
        float cx = pr[0], cy = pr[1], w = pr[2], h = pr[3];
        float px1 = cx - 0.5f * w, py1 = cy - 0.5f * h;
        float px2 = cx + 0.5f * w, py2 = cy + 0.5f * h;
        float v = iou_corner(gx1, gy1, gx2, gy2, areaG, px1, py1, px2, py2, w * h);
        if (v > best) { best = v; bidx = p; }   // strict > keeps lowest p within stride
    }
    // wave32 shuffle reduction (argmax, tie-break lower idx)
    #pragma unroll
    for (int off = 16; off > 0; off >>= 1) {
        float ov = __shfl_down(best, off);
        int   oi = __shfl_down(bidx, off);
        if (ov > best || (ov == best && oi < bidx)) { best = ov; bidx = oi; }
    }
    __shared__ float sv[8];
    __shared__ int   si[8];
    int lane = threadIdx.x & 31, wave = threadIdx.x >> 5;
    if (lane == 0) { sv[wave] = best; si[wave] = bidx; }
    __syncthreads();
    if (threadIdx.x == 0) {
        int nw = blockDim.x >> 5;
        for (int wv = 1; wv < nw; ++wv) {
            if (sv[wv] > best || (sv[wv] == best && si[wv] < bidx)) { best = sv[wv]; bidx = si[wv]; }
        }
        bp[blockIdx.x] = bidx;
    }
}

// One block per (b, tile of 256 priors): full match + repulsion loss contribution.
__global__ void k_loss(const float* __restrict__ pred_loc,  // [B,P,4]
                       const float* __restrict__ priors,    // [P,4] center-size
                       const float* __restrict__ gt,        // [B,G,5]
                       const int*   __restrict__ bp,        // [B,G]
                       float* __restrict__ acc) {
    const int tiles = (NPRIOR + PTILE - 1) / PTILE;
    int b    = blockIdx.x / tiles;
    int tile = blockIdx.x % tiles;
    int tid  = threadIdx.x;

    __shared__ float sgt[NGT * 5];
    __shared__ int   sbp[NGT];

    // ---- CDNA5 async global->LDS staging of GT rows + best-prior indices ----
    {
        const float* gsrc = gt + (size_t)b * NGT * 5;
        if (tid < NGT * 5) {
            unsigned lds_off = (unsigned)(unsigned long long)(const void*)&sgt[tid];
            unsigned long long ga = (unsigned long long)(const void*)(gsrc + tid);
            asm volatile("global_load_async_to_lds_b32 %0, %1, off"
                         :: "v"(lds_off), "v"(ga) : "memory");
        }
        if (tid < NGT) {
            unsigned lds_off = (unsigned)(unsigned long long)(const void*)&sbp[tid];
            unsigned long long ga = (unsigned long long)(const void*)(bp + (size_t)b * NGT + tid);
            asm volatile("global_load_async_to_lds_b32 %0, %1, off"
                         :: "v"(lds_off), "v"(ga) : "memory");
        }
        asm volatile("s_wait_asynccnt 0" ::: "memory");
    }
    __syncthreads();

    int p = tile * PTILE + tid;
    float lsum = 0.0f, pcnt = 0.0f;

    if (p < NPRIOR) {
        const float* pr = priors + (size_t)p * 4;
        float cx = pr[0], cy = pr[1], pw = pr[2], ph = pr[3];
        float px1 = cx - 0.5f * pw, py1 = cy - 0.5f * ph;
        float px2 = cx + 0.5f * pw, py2 = cy + 0.5f * ph;
        float areaP = pw * ph;

        // Pass 1: best GT per prior (argmax over g, first occurrence on ties)
        float best = -1.0f;
        int   bidx = 0;
        for (int g = 0; g < NGT; ++g) {
            float gx1 = sgt[g * 5 + 0], gy1 = sgt[g * 5 + 1];
            float gx2 = sgt[g * 5 + 2], gy2 = sgt[g * 5 + 3];
            float areaG = (gx2 - gx1) * (gy2 - gy1);
            float v = iou_corner(gx1, gy1, gx2, gy2, areaG, px1, py1, px2, py2, areaP);
            if (v > best) { best = v; bidx = g; }
        }
        // Sequential override (ensure each gt keeps its best prior; last write wins)
        for (int g = 0; g < NGT; ++g) {
            if (sbp[g] == p) { best = 2.0f; bidx = g; }
        }
        int conf = (best < THRESH) ? 0 : ((int)sgt[bidx * 5 + 4] + 1);

        // Pass 2: repulsion GT = argmax over g != bidx of ORIGINAL IoU (ov2 semantics)
        float rbest = -2.0f;
        int   ridx  = 0;
        for (int g = 0; g < NGT; ++g) {
            if (g == bidx) continue;
            float gx1 = sgt[g * 5 + 0], gy1 = sgt[g * 5 + 1];
            float gx2 = sgt[g * 5 + 2], gy2 = sgt[g * 5 + 3];
            float areaG = (gx2 - gx1) * (gy2 - gy1);
            float v = iou_corner(gx1, gy1, gx2, gy2, areaG, px1, py1, px2, py2, areaP);
            if (v > rbest) { rbest = v; ridx = g; }
        }
        float rx1 = sgt[ridx * 5 + 0], ry1 = sgt[ridx * 5 + 1];
        float rx2 = sgt[ridx * 5 + 2], ry2 = sgt[ridx * 5 + 3];

        // SSD decode of prediction
        const float* lc = pred_loc + ((size_t)b * NPRIOR + p) * 4;
        float dcx = cx + lc[0] * VAR0 * pw;
        float dcy = cy + lc[1] * VAR0 * ph;
        float dw  = pw * expf(lc[2] * VAR1);
        float dh  = ph * expf(lc[3] * VAR1);
        float dx1 = dcx - 0.5f * dw, dy1 = dcy - 0.5f * dh;
        float dx2 = dcx + 0.5f * dw, dy2 = dcy + 0.5f * dh;

        // IoG(rep_loc, decoded)
        float lx = fmaxf(rx1, dx1), ly = fmaxf(ry1, dy1);
        float hx = fminf(rx2, dx2), hy = fminf(ry2, dy2);
        float iw = fmaxf(hx - lx, 0.0f), ih = fmaxf(hy - ly, 0.0f);
        float inter = iw * ih;
        float areaR = (rx2 - rx1) * (ry2 - ry1);
        float iogv = inter / (areaR + EPSV);

        bool pos = conf > 0;
        if (pos) pcnt = 1.0f;
        if (pos && iogv < IOG_CAP) {
            if (iogv < SIGMA) lsum = -logf(1.0f - iogv + EPSV);
            else              lsum = (iogv - SIGMA) / (1.0f - SIGMA) - logf(1.0f - SIGMA);
        }
    }

    // Block reduction (wave32 shuffles + LDS across 8 waves)
    #pragma unroll
    for (int off = 16; off > 0; off >>= 1) {
        lsum += __shfl_down(lsum, off);
        pcnt += __shfl_down(pcnt, off);
    }
    __shared__ float sl[8], sc[8];
    int lane = tid & 31, wave = tid >> 5;
    if (lane == 0) { sl[wave] = lsum; sc[wave] = pcnt; }
    __syncthreads();
    if (tid == 0) {
        int nw = blockDim.x >> 5;
        for (int wv = 1; wv < nw; ++wv) { lsum += sl[wv]; pcnt += sc[wv]; }
        atomicAdd(&acc[0], lsum);
        atomicAdd(&acc[1], pcnt);
    }
}

__global__ void k_final(const float* __restrict__ acc, float* __restrict__ out) {
    if (threadIdx.x == 0) out[0] = acc[0] / acc[1];
}

extern "C" void kernel_launch(void* const* d_in, const int* in_sizes, int n_in,
                              void* d_out, int out_size, void* d_ws, size_t ws_size,
                              hipStream_t stream) {
    (void)in_sizes; (void)n_in; (void)out_size; (void)ws_size;
    const float* pred_loc = (const float*)d_in[0];  // [B,P,4] f32
    const float* priors   = (const float*)d_in[1];  // [P,4]   f32
    const float* gt       = (const float*)d_in[2];  // [B,G,5] f32
    float* out = (float*)d_out;

    float* acc = (float*)d_ws;
    int*   bpw = (int*)((char*)d_ws + 16);

    hipLaunchKernelGGL(k_init, dim3(1), dim3(32), 0, stream, acc);
    hipLaunchKernelGGL(k_best_prior, dim3(BATCH * NGT), dim3(256), 0, stream,
                       priors, gt, bpw);
    const int tiles = (NPRIOR + PTILE - 1) / PTILE;
    hipLaunchKernelGGL(k_loss, dim3(BATCH * tiles), dim3(256), 0, stream,
                       pred_loc, priors, gt, bpw, acc);
    hipLaunchKernelGGL(k_final, dim3(1), dim3(1), 0, stream, acc, out);
}